// DAMSM_6983616824261
// MI455X (gfx1250) — compile-verified
//
#include <hip/hip_runtime.h>
#include <math.h>

typedef __attribute__((ext_vector_type(2))) float v2f;
typedef __attribute__((ext_vector_type(8))) float v8f;

#define G1c 4.0f
#define G2c 5.0f
#define G3c 10.0f
#define LAMc 5.0f
#define EPSc 1e-8f

#define Bn 64
#define Pn 289      // H*W regions
#define Dn 256      // feature dim
#define Tn 24       // words
#define PPn 304     // P padded to multiple of 16
#define WSTR 260    // LDS row stride (floats) for word / rc tiles (bank-friendly, 8B aligned)
#define SSTR 34     // LDS row stride (floats) for S / attn buffer

// ---------------------------------------------------------------------------
// Kernel A: one workgroup per (word-sample i, image-sample j) pair.
// Computes sim[j*64+i] = G3 * logsumexp_t( G2 * cos(word_i[t], rc_ij[t]) )
// ---------------------------------------------------------------------------
__global__ __launch_bounds__(256)
void damsm_word_kernel(const float* __restrict__ image,
                       const float* __restrict__ word,
                       float* __restrict__ simW)
{
    __shared__ float WiL[32 * WSTR];     // word_i, rows 24..31 zero
    __shared__ float SmL[PPn * SSTR];    // S -> a1 -> a2 (attn) buffer
    __shared__ float rcL[32 * WSTR];     // region context per word
    __shared__ float redA[32 * 8], redB[32 * 8], redC[32 * 8];
    __shared__ float colmax[32], colsum[32], cosb[32];

    const int tid  = threadIdx.x;
    const int wave = tid >> 5;
    const int lane = tid & 31;
    const int lo   = lane & 15;     // N / M-minor index within half-wave
    const int hi   = lane >> 4;     // 0: lanes 0-15, 1: lanes 16-31
    const int kb   = hi << 1;       // K base for A/B operands: 0 or 2

    const int bid = blockIdx.x;
    const int i = bid & 63;         // word sample
    const int j = bid >> 6;         // image sample

    const float* __restrict__ ctx = image + (size_t)j * Pn * Dn;
    const float* __restrict__ wv  = word  + (size_t)i * Tn * Dn;

    // ---- stage word_i into LDS (pad rows 24..31 with zeros) ----
    for (int idx = tid; idx < 32 * Dn; idx += 256) {
        int t = idx >> 8, d = idx & 255;
        WiL[t * WSTR + d] = (t < Tn) ? wv[t * Dn + d] : 0.0f;
    }
    __syncthreads();

    // ---- step 1: S[p,t] = sum_d ctx[p,d] * Wi[t,d]  (289x24) via WMMA f32 16x16x4 ----
    for (int job = wave; job < 19 * 2; job += 8) {
        const int p0 = (job >> 1) * 16;
        const int t0 = (job & 1) * 16;
        const int arow = min(p0 + lo, Pn - 1);                 // clamp OOB rows (coeffs masked later)
        const float* aptr = ctx + (size_t)arow * Dn + kb;      // A[m][kb..kb+1]
        const float* bptr = WiL + (t0 + lo) * WSTR + kb;       // B[kb..kb+1][n]
        v8f acc0 = {}, acc1 = {};
        for (int k = 0; k < Dn; k += 8) {
            float2 af0 = *(const float2*)(aptr + k);
            float2 bf0 = *(const float2*)(bptr + k);
            float2 af1 = *(const float2*)(aptr + k + 4);
            float2 bf1 = *(const float2*)(bptr + k + 4);
            v2f a0, b0, a1, b1;
            a0.x = af0.x; a0.y = af0.y; b0.x = bf0.x; b0.y = bf0.y;
            a1.x = af1.x; a1.y = af1.y; b1.x = bf1.x; b1.y = bf1.y;
            acc0 = __builtin_amdgcn_wmma_f32_16x16x4_f32(false, a0, false, b0,
                                                         (short)0, acc0, false, false);
            acc1 = __builtin_amdgcn_wmma_f32_16x16x4_f32(false, a1, false, b1,
                                                         (short)0, acc1, false, false);
        }
        v8f acc = acc0 + acc1;
#pragma unroll
        for (int r = 0; r < 8; ++r) {
            int pr = p0 + r + (hi << 3);
            if (pr < Pn) SmL[pr * SSTR + t0 + lo] = acc[r];
        }
    }
    __syncthreads();

    // ---- step 2: softmax over t per region p (t < 24) ----
    for (int p = tid; p < Pn; p += 256) {
        float vals[Tn];
        float m = -3.0e38f;
        const int base = p * SSTR;
#pragma unroll
        for (int t = 0; t < Tn; ++t) { float v = SmL[base + t]; vals[t] = v; m = fmaxf(m, v); }
        float s = 0.0f;
#pragma unroll
        for (int t = 0; t < Tn; ++t) { float e = expf(vals[t] - m); vals[t] = e; s += e; }
        float inv = 1.0f / s;
#pragma unroll
        for (int t = 0; t < Tn; ++t) SmL[base + t] = vals[t] * inv;
    }
    // zero pad rows 289..303 so the K dimension of step 4 is clean
    for (int idx = tid; idx < (PPn - Pn) * 32; idx += 256) {
        int p = Pn + (idx >> 5), t = idx & 31;
        SmL[p * SSTR + t] = 0.0f;
    }
    __syncthreads();

    // ---- step 3: softmax over p of G1*a1, per word column t (chunked reduction) ----
    {
        const int tcol = tid & 31;
        const int chn  = tid >> 5;                  // 8 chunks of <=37 regions
        const int pLo  = chn * 37;
        const int pHi  = min(Pn, pLo + 37);
        const bool act = (tcol < Tn);
        float lm = -3.0e38f;
        if (act) for (int p = pLo; p < pHi; ++p) lm = fmaxf(lm, G1c * SmL[p * SSTR + tcol]);
        redA[tcol * 8 + chn] = lm;
        __syncthreads();
        if (tid < 32) {
            float m = -3.0e38f;
            for (int c = 0; c < 8; ++c) m = fmaxf(m, redA[tid * 8 + c]);
            colmax[tid] = m;
        }
        __syncthreads();
        float ls = 0.0f;
        if (act) {
            float m = colmax[tcol];
            for (int p = pLo; p < pHi; ++p) ls += expf(G1c * SmL[p * SSTR + tcol] - m);
        }
        redA[tcol * 8 + chn] = ls;
        __syncthreads();
        if (tid < 32) {
            float s = 0.0f;
            for (int c = 0; c < 8; ++c) s += redA[tid * 8 + c];
            colsum[tid] = s;
        }
        __syncthreads();
        if (act) {
            float m = colmax[tcol], inv = 1.0f / colsum[tcol];
            for (int p = pLo; p < pHi; ++p) {
                float v = SmL[p * SSTR + tcol];
                SmL[p * SSTR + tcol] = expf(G1c * v - m) * inv;
            }
        }
        __syncthreads();
    }

    // ---- step 4: rc[t,d] = sum_p a2[p,t] * ctx[p,d]  (M=t pad32, N=d, K=p pad304) ----
    for (int job = wave; job < 2 * 16; job += 8) {
        const int t0 = (job & 1) * 16;
        const int d0 = (job >> 1) * 16;
        v8f acc0 = {}, acc1 = {};
        for (int k = 0; k < PPn; k += 8) {
            v2f a0, b0, a1, b1;
            // A[m][kk] = a2[k+kk][t0+m]
            a0.x = SmL[(k + kb) * SSTR + t0 + lo];
            a0.y = SmL[(k + kb + 1) * SSTR + t0 + lo];
            a1.x = SmL[(k + 4 + kb) * SSTR + t0 + lo];
            a1.y = SmL[(k + 4 + kb + 1) * SSTR + t0 + lo];
            // B[kk][n] = ctx[k+kk][d0+n]  (clamped rows multiply zero coeffs)
            int r00 = min(k + kb, Pn - 1),     r01 = min(k + kb + 1, Pn - 1);
            int r10 = min(k + 4 + kb, Pn - 1), r11 = min(k + 4 + kb + 1, Pn - 1);
            b0.x = ctx[(size_t)r00 * Dn + d0 + lo];
            b0.y = ctx[(size_t)r01 * Dn + d0 + lo];
            b1.x = ctx[(size_t)r10 * Dn + d0 + lo];
            b1.y = ctx[(size_t)r11 * Dn + d0 + lo];
            acc0 = __builtin_amdgcn_wmma_f32_16x16x4_f32(false, a0, false, b0,
                                                         (short)0, acc0, false, false);
            acc1 = __builtin_amdgcn_wmma_f32_16x16x4_f32(false, a1, false, b1,
                                                         (short)0, acc1, false, false);
        }
        v8f acc = acc0 + acc1;
#pragma unroll
        for (int r = 0; r < 8; ++r) {
            int tr = t0 + r + (hi << 3);
            rcL[tr * WSTR + d0 + lo] = acc[r];
        }
    }
    __syncthreads();

    // ---- step 5: per-word cosine, then logsumexp -> sim[j,i] ----
    {
        const int t = tid & 31;
        const int c = tid >> 5;
        float sn = 0.0f, sr = 0.0f, sw = 0.0f;
        if (t < Tn) {
            const int dLo = c * 32;
            for (int d = dLo; d < dLo + 32; ++d) {
                float w = WiL[t * WSTR + d];
                float r = rcL[t * WSTR + d];
                sn += w * r; sr += r * r; sw += w * w;
            }
        }
        redA[t * 8 + c] = sn;
        redB[t * 8 + c] = sr;
        redC[t * 8 + c] = sw;
        __syncthreads();
        if (tid < Tn) {
            float n = 0.0f, r2 = 0.0f, w2 = 0.0f;
            for (int cc = 0; cc < 8; ++cc) {
                n  += redA[tid * 8 + cc];
                r2 += redB[tid * 8 + cc];
                w2 += redC[tid * 8 + cc];
            }
            cosb[tid] = n / (sqrtf(w2) * sqrtf(r2) + EPSc);
        }
        __syncthreads();
        if (tid == 0) {
            float m = -3.0e38f;
            for (int t2 = 0; t2 < Tn; ++t2) m = fmaxf(m, G2c * cosb[t2]);
            float s = 0.0f;
            for (int t2 = 0; t2 < Tn; ++t2) s += expf(G2c * cosb[t2] - m);
            simW[j * Bn + i] = G3c * (m + logf(s));
        }
    }
}

// ---------------------------------------------------------------------------
// Kernel B: sentence cosine-similarity matrix sc[x,y], scaled by G3
// ---------------------------------------------------------------------------
__global__ __launch_bounds__(256)
void damsm_sent_kernel(const float* __restrict__ g, const float* __restrict__ s,
                       float* __restrict__ simS)
{
    int gid = blockIdx.x * 256 + threadIdx.x;   // 4096 = 64*64
    int x = gid >> 6, y = gid & 63;
    const float* gp = g + x * Dn;
    const float* sp = s + y * Dn;
    float dot = 0.0f, ng = 0.0f, ns = 0.0f;
    for (int d = 0; d < Dn; ++d) {
        float a = gp[d], b = sp[d];
        dot += a * b; ng += a * a; ns += b * b;
    }
    simS[x * Bn + y] = dot / fmaxf(sqrtf(ng) * sqrtf(ns), EPSc) * G3c;
}

// ---------------------------------------------------------------------------
// Kernel C: masked bidirectional CE for both matrices -> scalar loss
// ---------------------------------------------------------------------------
__device__ __forceinline__ float ce_line(const float* __restrict__ M, int r,
                                         const int* cl, bool rowmode)
{
    const int cr = cl[r];
    float m = -3.0e38f;
    for (int c = 0; c < Bn; ++c) {
        float v = rowmode ? M[r * Bn + c] : M[c * Bn + r];
        if (cl[c] == cr && c != r) v = -3.0e38f;   // mask same-class negatives
        m = fmaxf(m, v);
    }
    float s = 0.0f;
    for (int c = 0; c < Bn; ++c) {
        float v = rowmode ? M[r * Bn + c] : M[c * Bn + r];
        if (cl[c] == cr && c != r) v = -3.0e38f;
        s += expf(v - m);
    }
    float diag = M[r * (Bn + 1)];
    return -(diag - (m + logf(s)));
}

__global__ __launch_bounds__(64)
void damsm_ce_kernel(const float* __restrict__ simW, const float* __restrict__ simS,
                     const int* __restrict__ class_ids, float* __restrict__ out)
{
    __shared__ int   cl[Bn];
    __shared__ float part[Bn];
    int tid = threadIdx.x;
    cl[tid] = class_ids[tid];
    __syncthreads();
    float w = ce_line(simW, tid, cl, true) + ce_line(simW, tid, cl, false);
    float s = ce_line(simS, tid, cl, true) + ce_line(simS, tid, cl, false);
    part[tid] = w * (LAMc / (float)Bn) + s * (1.0f / (LAMc * (float)Bn));
    __syncthreads();
    if (tid == 0) {
        float tot = 0.0f;
        for (int k = 0; k < Bn; ++k) tot += part[k];
        out[0] = tot;
    }
}

extern "C" void kernel_launch(void* const* d_in, const int* in_sizes, int n_in,
                              void* d_out, int out_size, void* d_ws, size_t ws_size,
                              hipStream_t stream)
{
    (void)in_sizes; (void)n_in; (void)out_size; (void)ws_size;
    const float* image = (const float*)d_in[0];   // [64,17,17,256]
    const float* word  = (const float*)d_in[1];   // [64,24,256]
    const float* gvec  = (const float*)d_in[2];   // [64,256]
    const float* svec  = (const float*)d_in[3];   // [64,256]
    const int*   cls   = (const int*)d_in[4];     // [64]

    float* simW = (float*)d_ws;                   // [64*64], indexed [j,i]
    float* simS = simW + Bn * Bn;                 // [64*64], indexed [x,y]

    damsm_word_kernel<<<dim3(Bn * Bn), dim3(256), 0, stream>>>(image, word, simW);
    damsm_sent_kernel<<<dim3(16), dim3(256), 0, stream>>>(gvec, svec, simS);
    damsm_ce_kernel<<<dim3(1), dim3(64), 0, stream>>>(simW, simS, cls, (float*)d_out);
}